// ROI_align_25D_79164837200562
// MI455X (gfx1250) — compile-verified
//
#include <hip/hip_runtime.h>

// Problem constants (match reference)
#define Bn  16
#define Cn  64
#define Hn  128
#define Wn  128
#define Dn  8
#define CSZ 16              // crop half-size = H * 1/8
#define NROI (Bn * 5)       // 80
#define OUT_TOT (NROI * Cn * 2 * 2 * Dn)   // 163840 floats

typedef float v2f __attribute__((ext_vector_type(2)));
typedef float v4f __attribute__((ext_vector_type(4)));
typedef float v8f __attribute__((ext_vector_type(8)));

#define KSTR 520            // padded LDS row stride (floats); 2080B rows stagger K-half banks, 16B aligned
#define WPAD 40             // padded weight-table row (span <= 34)
#define PFD  6              // async pipeline depth (chunks in flight)
#define NBUF 8              // LDS ring buffers (> PFD), power of two; +1 dummy
#define SLICES 4            // row-slices per ROI (when workspace is big enough)

#define STR2(x) #x
#define STR(x) STR2(x)
#define PFD_M1 5            // PFD-1, immediate for s_wait_asynccnt

__global__ __launch_bounds__(512)
void roi_align_25d_wmma(const float* __restrict__ fea,
                        const float* __restrict__ kp,
                        float* __restrict__ dst)   // out (S==1) or ws partials (S>1)
{
    __shared__ __align__(16) float stage[NBUF + 1][4 * KSTR]; // ring + 1 dummy buffer
    __shared__ float WYs[2 * WPAD];    // [ph][row]
    __shared__ float WXs[2 * WPAD];    // [pw][col]

    const int n     = blockIdx.x;      // ROI id
    const int slice = blockIdx.y;
    const int S     = gridDim.y;
    const int bimg  = n / 5;
    const int tid   = threadIdx.x;

    // ---- ROI parameters (block-uniform) ----
    const float kx   = kp[n * 3 + 0] * (float)Wn;
    const float ky   = kp[n * 3 + 1] * (float)Hn;
    const float xmin = fminf(fmaxf(kx - (float)CSZ, 0.f), (float)(Wn - 1));
    const float xmax = fminf(fmaxf(kx + (float)CSZ, 0.f), (float)(Wn - 1));
    const float ymin = fminf(fmaxf(ky - (float)CSZ, 0.f), (float)(Hn - 1));
    const float ymax = fminf(fmaxf(ky + (float)CSZ, 0.f), (float)(Hn - 1));
    const float roi_w = fmaxf(xmax - xmin, 1.f);
    const float roi_h = fmaxf(ymax - ymin, 1.f);
    const float bin_w = roi_w * 0.5f;
    const float bin_h = roi_h * 0.5f;
    const int   gw    = (int)ceilf(roi_w * 0.5f);
    const int   gh    = (int)ceilf(roi_h * 0.5f);
    const float count = (float)(gh * gw);

    const int x0 = (int)floorf(xmin);
    const int y0 = (int)floorf(ymin);
    const int x1 = min(Wn - 1, (int)floorf(xmin + roi_w) + 1);
    const int y1 = min(Hn - 1, (int)floorf(ymin + roi_h) + 1);
    const int span_x = x1 - x0 + 1;        // <= 34
    const int span_y = y1 - y0 + 1;        // <= 34
    const int nq     = (span_x + 3) >> 2;  // x-quads per row

    // row range of this slice (uniform)
    const int r0s = (span_y * slice) / S;
    const int r1s = (span_y * (slice + 1)) / S;
    const int nchunk = (r1s - r0s) * nq;

    // ---- zero weight tables ----
    if (tid < 2 * WPAD) { WYs[tid] = 0.f; WXs[tid] = 0.f; }
    __syncthreads();

    // ---- per-axis accumulated bilinear weights: one (axis, ph, index) per thread ----
    {
        int t = tid;
        if (t < 2 * span_y) {
            int ph = (t >= span_y) ? 1 : 0;
            int r  = t - ph * span_y;
            int iy = y0 + r;
            float step = bin_h / (float)gh;
            float w = 0.f;
            for (int g = 0; g < gh; ++g) {
                float y  = ymin + (float)ph * bin_h + ((float)g + 0.5f) * step;
                float c0 = fmaxf(y, 0.f);
                int   l0 = (int)floorf(c0);
                int lo, hi; float fr;
                if (l0 >= Hn - 1) { lo = Hn - 1; hi = Hn - 1; fr = 0.f; }
                else              { lo = l0;     hi = l0 + 1; fr = c0 - (float)l0; }
                if (lo == iy) w += 1.f - fr;
                if (hi == iy) w += fr;
            }
            WYs[ph * WPAD + r] = w;
        } else if (t < 2 * span_y + 2 * span_x) {
            int u  = t - 2 * span_y;
            int pw = (u >= span_x) ? 1 : 0;
            int cx = u - pw * span_x;
            int ix = x0 + cx;
            float step = bin_w / (float)gw;
            float w = 0.f;
            for (int g = 0; g < gw; ++g) {
                float x  = xmin + (float)pw * bin_w + ((float)g + 0.5f) * step;
                float c0 = fmaxf(x, 0.f);
                int   l0 = (int)floorf(c0);
                int lo, hi; float fr;
                if (l0 >= Wn - 1) { lo = Wn - 1; hi = Wn - 1; fr = 0.f; }
                else              { lo = l0;     hi = l0 + 1; fr = c0 - (float)l0; }
                if (lo == ix) w += 1.f - fr;
                if (hi == ix) w += fr;
            }
            WXs[pw * WPAD + cx] = w;
        }
    }

    const int lane = tid & 31;
    const int wave = tid >> 5;
    const int m    = lane & 15;            // M row (ph,pw when m<4)
    const int kb   = (lane >> 4) << 1;     // K half: 0 or 2
    const int pwm  = m & 1;

    // per-thread staging constants: one 16B granule per thread per chunk
    const int cc = tid >> 3;               // channel 0..63
    const int kk = (tid >> 1) & 3;         // tap within quad
    const int d4 = (tid & 1) << 2;         // depth half
    const size_t fbase = (size_t)bimg * Cn * Hn * Wn * Dn;
    const size_t pbase = fbase + (size_t)cc * (Hn * Wn * Dn) + d4;

    // Issue one chunk's staging: async global->LDS b128 per thread (ASYNCcnt),
    // out-of-span taps write zeros. ck >= nchunk -> dummy prefetch (keeps the
    // per-wave outstanding ASYNCcnt invariant so the wait immediate is constant).
    auto issue = [&](int ck) {
        const bool real = (ck < nchunk);
        int buf, r, q;
        if (real) { buf = ck & (NBUF - 1); r = ck / nq; q = ck - r * nq; }
        else      { buf = NBUF;            r = 0;       q = 0; }
        const int iy = y0 + r0s + r;
        const int ix = x0 + (q << 2) + kk;
        float* lp = &stage[buf][kk * KSTR + cc * 8 + d4];
        if (ix <= x1) {
            const float* gp = fea + pbase + (size_t)iy * (Wn * Dn) + (size_t)ix * Dn;
            asm volatile("global_load_async_to_lds_b128 %0, %1, off"
                         :: "v"((unsigned)(size_t)lp),
                            "v"((unsigned long long)(size_t)gp)
                         : "memory");
        } else {
            *(v4f*)lp = (v4f){0.f, 0.f, 0.f, 0.f};
        }
    };

    v8f acc0 = {};
    v8f acc1 = {};

    if (nchunk > 0) {
        #pragma unroll
        for (int p = 0; p < PFD; ++p) issue(p);

        for (int ck = 0; ck < nchunk; ++ck) {
            // chunk ck's async loads have landed once each wave's outstanding <= PFD-1
            asm volatile("s_wait_asynccnt " STR(PFD_M1) ::: "memory");
            __syncthreads();   // make every wave's LDS writes for chunk ck visible

            const int bs = ck & (NBUF - 1);
            const int r  = ck / nq, q = ck - (ck / nq) * nq;
            const int ra = r0s + r;          // absolute row in weight table

            // A[m][k] = WY[ph][row] * WX[pw][col]; rows m>=4 are zero (free: HBM-bound)
            float wy  = (m < 2) ? WYs[ra] : ((m < 4) ? WYs[WPAD + ra] : 0.f);
            float wxa = WXs[pwm * WPAD + (q << 2) + kb];
            float wxb = WXs[pwm * WPAD + (q << 2) + kb + 1];
            v2f A; A.x = wy * wxa; A.y = wy * wxb;

            int ncol0 = (wave * 2) * 16 + m;
            v2f B0; B0.x = stage[bs][kb * KSTR + ncol0];
                    B0.y = stage[bs][(kb + 1) * KSTR + ncol0];
            acc0 = __builtin_amdgcn_wmma_f32_16x16x4_f32(false, A, false, B0,
                                                         (short)0, acc0, false, false);
            int ncol1 = ncol0 + 16;
            v2f B1; B1.x = stage[bs][kb * KSTR + ncol1];
                    B1.y = stage[bs][(kb + 1) * KSTR + ncol1];
            acc1 = __builtin_amdgcn_wmma_f32_16x16x4_f32(false, A, false, B1,
                                                         (short)0, acc1, false, false);

            issue(ck + PFD);   // writes buffer (ck-2)%NBUF: reads fenced 2 barriers ago
        }
    }

    // C/D layout: M rows 0..3 live in VGPRs 0..3 of lanes 0..15.
    if (lane < 16) {
        float* obase = dst + (size_t)slice * OUT_TOT + (size_t)n * (Cn * 2 * 2 * Dn);
        #pragma unroll
        for (int t = 0; t < 2; ++t) {
            int cd = (wave * 2 + t) * 16 + lane;
            int c = cd >> 3, d = cd & 7;
            v8f a = t ? acc1 : acc0;
            #pragma unroll
            for (int j = 0; j < 4; ++j) {    // j = ph*2+pw
                int ph = j >> 1, pw = j & 1;
                obase[c * 32 + ph * 16 + pw * 8 + d] = a[j] / count;
            }
        }
    }
}

// Deterministic fixed-order reduction of row-slice partials.
__global__ __launch_bounds__(256)
void roi_align_reduce(const float* __restrict__ ws, float* __restrict__ out, int S)
{
    int i = blockIdx.x * blockDim.x + threadIdx.x;
    if (i < OUT_TOT) {
        float s = 0.f;
        for (int k = 0; k < S; ++k) s += ws[(size_t)k * OUT_TOT + i];
        out[i] = s;
    }
}

extern "C" void kernel_launch(void* const* d_in, const int* in_sizes, int n_in,
                              void* d_out, int out_size, void* d_ws, size_t ws_size,
                              hipStream_t stream) {
    (void)in_sizes; (void)n_in; (void)out_size;
    const float* fea = (const float*)d_in[0];
    const float* kp  = (const float*)d_in[1];
    float*       out = (float*)d_out;

    const size_t need = (size_t)SLICES * OUT_TOT * sizeof(float);
    const int S = (d_ws != nullptr && ws_size >= need) ? SLICES : 1;
    float* dst = (S > 1) ? (float*)d_ws : out;

    roi_align_25d_wmma<<<dim3(NROI, S), dim3(512), 0, stream>>>(fea, kp, dst);
    if (S > 1)
        roi_align_reduce<<<dim3((OUT_TOT + 255) / 256), dim3(256), 0, stream>>>(
            (const float*)d_ws, out, S);
}